// MobileMambaModule_37615323578790
// MI455X (gfx1250) — compile-verified
//
#include <hip/hip_runtime.h>
#include <math.h>

#define DEVINL __device__ __forceinline__

typedef float v2f __attribute__((ext_vector_type(2)));
typedef float v8f __attribute__((ext_vector_type(8)));

static constexpr int Bn = 16, Ctot = 256, L = 4096;
static constexpr int Dm = 64;
static constexpr int CHUNK = 128, NCHUNK = 32;

// ---- workspace layout (float offsets), total ~30.8M floats (~123 MiB) ----
static constexpr size_t OFF_MEAN = 0;                                   // 1024
static constexpr size_t OFF_EFF0 = 2048;                                // 16*64*9
static constexpr size_t OFF_EFF1 = OFF_EFF0 + (size_t)16*64*9;          // 16*64*25
static constexpr size_t OFF_EFF2 = OFF_EFF1 + (size_t)16*64*25;         // 16*64*49
static constexpr size_t OFF_UP   = 131072;                              // B*64*L
static constexpr size_t OFF_XI   = OFF_UP   + (size_t)Bn*64*L;
static constexpr size_t OFF_Z    = OFF_XI   + (size_t)Bn*64*L;
static constexpr size_t OFF_XC   = OFF_Z    + (size_t)Bn*64*L;
static constexpr size_t OFF_UT   = OFF_XC   + (size_t)Bn*64*L;          // (b,l,d) transposed xc
static constexpr size_t OFF_XDBL = OFF_UT   + (size_t)Bn*64*L;          // (dir,b,l,8)
static constexpr size_t OFF_AGGA = OFF_XDBL + (size_t)2*Bn*L*8;
static constexpr size_t OFF_AGGB = OFF_AGGA + (size_t)2*Bn*NCHUNK*64;
static constexpr size_t OFF_H0   = OFF_AGGB + (size_t)2*Bn*NCHUNK*64;
static constexpr size_t OFF_Y    = OFF_H0   + (size_t)2*Bn*NCHUNK*64;   // (dir,b,l,d)

DEVINL float softplusf(float v) { return (v > 20.f) ? v : log1pf(__expf(v)); }
DEVINL float siluf(float v)     { return v / (1.f + __expf(-v)); }

DEVINL v8f wmma4(v2f a, v2f b, v8f c) {
#if defined(__gfx1250__)
  // D = A(16x4 f32) * B(4x16 f32) + C(16x16 f32)
  return __builtin_amdgcn_wmma_f32_16x16x4_f32(false, a, false, b, (short)0, c, false, false);
#else
  c[0] += a.x * b.x; // host stub (never executed)
  return c;
#endif
}

// ---------------- passthrough: channels 128..255 ----------------
__global__ __launch_bounds__(256) void k_copy(const float4* __restrict__ x4,
                                              float4* __restrict__ o4) {
  size_t idx = (size_t)blockIdx.x * 256 + threadIdx.x;   // 2,097,152 total
  size_t per_b = (size_t)128 * L / 4;
  size_t b = idx / per_b, r = idx % per_b;
  size_t off = ((size_t)b * Ctot + 128) * (L / 4) + r;
  o4[off] = x4[off];
}

// ---------------- per-channel mean of xl (channels 64..127) ----------------
__global__ __launch_bounds__(256) void k_mean(const float* __restrict__ x, float* __restrict__ ws) {
  int b = blockIdx.x >> 6, c = blockIdx.x & 63;
  const float* p = x + ((size_t)(b * Ctot + 64 + c)) * L;
  float s = 0.f;
  for (int i = threadIdx.x; i < L; i += 256) s += p[i];
  __shared__ float red[256];
  red[threadIdx.x] = s; __syncthreads();
  for (int st = 128; st > 0; st >>= 1) {
    if (threadIdx.x < st) red[threadIdx.x] += red[threadIdx.x + st];
    __syncthreads();
  }
  if (threadIdx.x == 0) ws[OFF_MEAN + b * 64 + c] = red[0] * (1.f / 4096.f);
}

// ---------------- gates (softmax) + effective dwconv kernels ----------------
__global__ __launch_bounds__(64) void k_gates_eff(float* __restrict__ ws,
    const float* gw0, const float* gw1, const float* gw2,
    const float* ew0, const float* ew1, const float* ew2) {
  int j = blockIdx.x >> 4, b = blockIdx.x & 15;
  const float* gw = (j == 0) ? gw0 : (j == 1 ? gw1 : gw2);
  const float* ew = (j == 0) ? ew0 : (j == 1 ? ew1 : ew2);
  float* eff = ws + ((j == 0) ? OFF_EFF0 : (j == 1 ? OFF_EFF1 : OFF_EFF2));
  int ks = (j == 0) ? 3 : (j == 1 ? 5 : 7);
  int k2 = ks * ks;
  __shared__ float lg[4], g[4];
  int t = threadIdx.x;
  if (t < 4) {
    float s = 0.f;
    for (int c = 0; c < 64; ++c) s += ws[OFF_MEAN + b * 64 + c] * gw[t * 64 + c];
    lg[t] = s;
  }
  __syncthreads();
  if (t == 0) {
    float m = fmaxf(fmaxf(lg[0], lg[1]), fmaxf(lg[2], lg[3]));
    float e0 = __expf(lg[0] - m), e1 = __expf(lg[1] - m);
    float e2 = __expf(lg[2] - m), e3 = __expf(lg[3] - m);
    float inv = 1.f / (e0 + e1 + e2 + e3);
    g[0] = e0 * inv; g[1] = e1 * inv; g[2] = e2 * inv; g[3] = e3 * inv;
  }
  __syncthreads();
  int c = t; // 64 threads
  for (int kk = 0; kk < k2; ++kk) {
    float s = 0.f;
    for (int e = 0; e < 4; ++e) s += g[e] * ew[((size_t)(e * 64 + c)) * k2 + kk];
    eff[((size_t)(b * 64 + c)) * k2 + kk] = s;
  }
}

// ---------------- local branch: 3 dwconvs + bn/mish/pw/bn, summed ----------------
template <int K>
DEVINL float dconv(const float* tile, int h, int w, const float* wgt) {
  constexpr int P = K / 2;
  float s = 0.f;
#pragma unroll
  for (int dy = 0; dy < K; ++dy)
#pragma unroll
    for (int dx = 0; dx < K; ++dx)
      s += tile[(h + 3 - P + dy) * 70 + (w + 3 - P + dx)] * wgt[dy * K + dx];
  return s;
}
DEVINL float branchf(float v, float s1, float b1, float pw, float s2, float b2) {
  v = v * s1 + b1;
  v = v * tanhf(softplusf(v));
  v *= pw;
  return v * s2 + b2;
}

__global__ __launch_bounds__(256) void k_local(const float* __restrict__ x,
    const float* __restrict__ ws,
    const float* g0a, const float* b0a, const float* p0a, const float* g0b, const float* b0b,
    const float* g1a, const float* b1a, const float* p1a, const float* g1b, const float* b1b,
    const float* g2a, const float* b2a, const float* p2a, const float* g2b, const float* b2b,
    float* __restrict__ out) {
  __shared__ float tile[70 * 70];
  __shared__ float eff[96];
  int b = blockIdx.x >> 6, c = blockIdx.x & 63, t = threadIdx.x;
  for (int i = t; i < 70 * 70; i += 256) tile[i] = 0.f;
  __syncthreads();
  const float* src = x + ((size_t)(b * Ctot + 64 + c)) * L;
  for (int i = t; i < L; i += 256) tile[((i >> 6) + 3) * 70 + (i & 63) + 3] = src[i];
  if (t < 9)       eff[t]      = ws[OFF_EFF0 + (size_t)(b * 64 + c) * 9 + t];
  else if (t < 34) eff[t]      = ws[OFF_EFF1 + (size_t)(b * 64 + c) * 25 + (t - 9)];
  else if (t < 83) eff[t]      = ws[OFF_EFF2 + (size_t)(b * 64 + c) * 49 + (t - 34)];
  __syncthreads();
  const float RSQ = rsqrtf(1.f + 1e-5f);
  float s1x = g0a[c] * RSQ, b1x = b0a[c], px = p0a[c], s2x = g0b[c] * RSQ, b2x = b0b[c];
  float s1y = g1a[c] * RSQ, b1y = b1a[c], py = p1a[c], s2y = g1b[c] * RSQ, b2y = b1b[c];
  float s1z = g2a[c] * RSQ, b1z = b2a[c], pz = p2a[c], s2z = g2b[c] * RSQ, b2z = b2b[c];
  float* dst = out + ((size_t)(b * Ctot + 64 + c)) * L;
  for (int i = t; i < L; i += 256) {
    int h = i >> 6, w = i & 63;
    float v0 = dconv<3>(tile, h, w, &eff[0]);
    float v1 = dconv<5>(tile, h, w, &eff[9]);
    float v2 = dconv<7>(tile, h, w, &eff[34]);
    dst[i] = branchf(v0, s1x, b1x, px, s2x, b2x) +
             branchf(v1, s1y, b1y, py, s2y, b2y) +
             branchf(v2, s1z, b1z, pz, s2z, b2z);
  }
}

// ---------------- Haar wavelet branch -> WS_UP ----------------
__global__ __launch_bounds__(256) void k_wavelet(const float* __restrict__ x,
    const float* __restrict__ wav_w, const float* __restrict__ wav_b,
    const float* __restrict__ wav_scale, float* __restrict__ ws) {
  __shared__ float xt[64 * 64];
  __shared__ float sub[4][34 * 34];
  int b = blockIdx.x >> 6, c = blockIdx.x & 63, t = threadIdx.x;
  const float* src = x + ((size_t)(b * Ctot + c)) * L;
  for (int i = t; i < 4096; i += 256) xt[i] = src[i];
  for (int i = t; i < 4 * 34 * 34; i += 256) (&sub[0][0])[i] = 0.f;
  __syncthreads();
  for (int i = t; i < 1024; i += 256) {
    int h = i >> 5, w = i & 31;
    float x00 = xt[(2 * h) * 64 + 2 * w],     x01 = xt[(2 * h) * 64 + 2 * w + 1];
    float x10 = xt[(2 * h + 1) * 64 + 2 * w], x11 = xt[(2 * h + 1) * 64 + 2 * w + 1];
    sub[0][(h + 1) * 34 + w + 1] = 0.5f * (x00 + x01 + x10 + x11);
    sub[1][(h + 1) * 34 + w + 1] = 0.5f * (x00 + x01 - x10 - x11);
    sub[2][(h + 1) * 34 + w + 1] = 0.5f * (x00 - x01 + x10 - x11);
    sub[3][(h + 1) * 34 + w + 1] = 0.5f * (x00 - x01 - x10 + x11);
  }
  __syncthreads();
  float* up = ws + OFF_UP + ((size_t)(b * 64 + c)) * L;
  for (int i = t; i < 1024; i += 256) {
    int h = i >> 5, w = i & 31;
    float tg[4];
#pragma unroll
    for (int k = 0; k < 4; ++k) {
      int ch = c * 4 + k;
      float s = wav_b[ch];
#pragma unroll
      for (int dy = 0; dy < 3; ++dy)
#pragma unroll
        for (int dx = 0; dx < 3; ++dx)
          s += sub[k][(h + dy) * 34 + (w + dx)] * wav_w[ch * 9 + dy * 3 + dx];
      tg[k] = s * wav_scale[ch];
    }
#pragma unroll
    for (int p = 0; p < 2; ++p)
#pragma unroll
      for (int q = 0; q < 2; ++q) {
        float sp = p ? 1.f : -1.f, sq = q ? 1.f : -1.f;
        up[(2 * h + p) * 64 + (2 * w + q)] =
            0.5f * (tg[0] + sp * tg[1] + sq * tg[2] + sp * sq * tg[3]);
      }
  }
}

// ---------------- WMMA: in-projection xz = in_w(128x64) @ xg ----------------
__global__ __launch_bounds__(64) void k_proj_in(const float* __restrict__ x,
    const float* __restrict__ in_w, float* __restrict__ ws) {
  __shared__ float s_w[128 * 64];
  __shared__ float s_xT[16 * 66];   // [n][c], stride 66 (even -> b64 loads)
  float* xi = ws + OFF_XI;
  float* zz = ws + OFF_Z;
  int t = threadIdx.x;
  int b = blockIdx.x >> 5, tb = blockIdx.x & 31;
  for (int i = t; i < 128 * 64; i += 64) s_w[i] = in_w[i];
  int lane = t & 31, wave = t >> 5;
  int n = lane & 15, koff = (lane >> 4) << 1, m = lane & 15, rhi = (lane >> 4) << 3;
  for (int tile = 0; tile < 8; ++tile) {
    int l0 = tb * 128 + tile * 16;
    __syncthreads();
    for (int i = 0; i < 16; ++i) {
      int c = i * 4 + (t >> 4), nn = t & 15;
      s_xT[nn * 66 + c] = x[((size_t)(b * Ctot + c)) * L + l0 + nn];
    }
    __syncthreads();
    v8f a0 = {}, a1 = {}, a2 = {}, a3 = {};
#pragma unroll
    for (int k = 0; k < 16; ++k) {
      int kk = k * 4 + koff;
      v2f bf = *(const v2f*)&s_xT[n * 66 + kk];
      v2f w0 = *(const v2f*)&s_w[((wave * 4 + 0) * 16 + m) * 64 + kk];
      v2f w1 = *(const v2f*)&s_w[((wave * 4 + 1) * 16 + m) * 64 + kk];
      v2f w2 = *(const v2f*)&s_w[((wave * 4 + 2) * 16 + m) * 64 + kk];
      v2f w3 = *(const v2f*)&s_w[((wave * 4 + 3) * 16 + m) * 64 + kk];
      a0 = wmma4(w0, bf, a0); a1 = wmma4(w1, bf, a1);
      a2 = wmma4(w2, bf, a2); a3 = wmma4(w3, bf, a3);
    }
    int l = l0 + n;
    float* dst = wave ? zz : xi;   // wave0 -> o in [0,64), wave1 -> [64,128)
#pragma unroll
    for (int mt = 0; mt < 4; ++mt) {
      v8f a = (mt == 0) ? a0 : (mt == 1) ? a1 : (mt == 2) ? a2 : a3;
      int ob = (wave * 4 + mt) * 16;
#pragma unroll
      for (int r = 0; r < 8; ++r) {
        int o = (ob + r + rhi) & 63;
        dst[((size_t)(b * 64 + o)) * L + l] = a[r];
      }
    }
  }
}

// ---------------- xc = silu(dwconv3x3(xi) + cb) ----------------
__global__ __launch_bounds__(256) void k_conv_xc(const float* __restrict__ cw,
    const float* __restrict__ cb, float* __restrict__ ws) {
  __shared__ float tile[66 * 66];
  __shared__ float wl[9];
  int b = blockIdx.x >> 6, c = blockIdx.x & 63, t = threadIdx.x;
  for (int i = t; i < 66 * 66; i += 256) tile[i] = 0.f;
  __syncthreads();
  const float* src = ws + OFF_XI + ((size_t)(b * 64 + c)) * L;
  for (int i = t; i < L; i += 256) tile[((i >> 6) + 1) * 66 + (i & 63) + 1] = src[i];
  if (t < 9) wl[t] = cw[c * 9 + t];
  __syncthreads();
  float bias = cb[c];
  float* dst = ws + OFF_XC + ((size_t)(b * 64 + c)) * L;
  for (int i = t; i < L; i += 256) {
    int h = i >> 6, w = i & 63;
    float s = bias;
#pragma unroll
    for (int dy = 0; dy < 3; ++dy)
#pragma unroll
      for (int dx = 0; dx < 3; ++dx)
        s += tile[(h + dy) * 66 + (w + dx)] * wl[dy * 3 + dx];
    dst[i] = siluf(s);
  }
}

// ------- LDS-tiled transpose xc -> ut (b,l,d) + xdbl projections (both dirs) -------
__global__ __launch_bounds__(256) void k_transpose0(const float* __restrict__ xpw,
                                                    float* __restrict__ ws) {
  __shared__ float tile[64 * 65];
  const float* xc = ws + OFF_XC;
  float* ut = ws + OFF_UT;
  float* xdbl = ws + OFF_XDBL;
  int b = blockIdx.x >> 6, lc = blockIdx.x & 63, t = threadIdx.x;
  int l0 = lc * 64;
  for (int p = 0; p < 16; ++p) {
    int d = p * 4 + (t >> 6), ll = t & 63;
    tile[d * 65 + ll] = xc[((size_t)(b * Dm + d)) * L + l0 + ll];
  }
  __syncthreads();
  for (int p = 0; p < 16; ++p) {
    int ll = p * 4 + (t >> 6), d = t & 63;
    ut[((size_t)b * L + l0 + ll) * 64 + d] = tile[d * 65 + ll];
  }
  if (t < 64) {
    int ll = t, l = l0 + ll;
    float s0[6] = {0, 0, 0, 0, 0, 0}, s1[6] = {0, 0, 0, 0, 0, 0};
    for (int d = 0; d < 64; ++d) {
      float u = tile[d * 65 + ll];
#pragma unroll
      for (int e = 0; e < 6; ++e) {
        s0[e] += xpw[e * 64 + d] * u;
        s1[e] += xpw[384 + e * 64 + d] * u;
      }
    }
    size_t i0 = ((size_t)b * L + l) * 8;
    size_t i1 = ((size_t)(16 + b) * L + ((l & 63) * 64 + (l >> 6))) * 8;
#pragma unroll
    for (int e = 0; e < 6; ++e) { xdbl[i0 + e] = s0[e]; xdbl[i1 + e] = s1[e]; }
  }
}

// ---------------- chunked parallel selective scan ----------------
__global__ __launch_bounds__(64) void k_scan1(const float* __restrict__ dtw,
    const float* __restrict__ dtb, const float* __restrict__ alog, float* __restrict__ ws) {
  int blk = blockIdx.x;
  int dir = blk >> 9, b = (blk >> 5) & 15, ch = blk & 31;
  int d = threadIdx.x;
  const float* ut = ws + OFF_UT;
  const float* xd = ws + OFF_XDBL + ((size_t)(dir * 16 + b) * L) * 8;
  float4 w = *(const float4*)&dtw[(dir * 64 + d) * 4];
  float bia = dtb[dir * 64 + d];
  float A = -__expf(alog[dir * 64 + d]);
  float aP = 1.f, bP = 0.f;
  for (int i = 0; i < CHUNK; ++i) {
    int l = ch * CHUNK + i;
    int row = dir ? ((l & 63) * 64 + (l >> 6)) : l;
    float u = ut[((size_t)b * L + row) * 64 + d];
    const float* p = xd + (size_t)l * 8;
    float delta = softplusf(p[0] * w.x + p[1] * w.y + p[2] * w.z + p[3] * w.w + bia);
    float a = __expf(delta * A);
    float bb = delta * p[4] * u;
    aP *= a;
    bP = a * bP + bb;
  }
  size_t o = ((size_t)(dir * 16 + b) * NCHUNK + ch) * 64 + d;
  ws[OFF_AGGA + o] = aP;
  ws[OFF_AGGB + o] = bP;
}

__global__ __launch_bounds__(64) void k_scan2(float* __restrict__ ws) {
  int g = blockIdx.x, d = threadIdx.x;
  float h = 0.f;
  for (int ch = 0; ch < NCHUNK; ++ch) {
    size_t o = ((size_t)g * NCHUNK + ch) * 64 + d;
    ws[OFF_H0 + o] = h;
    h = ws[OFF_AGGA + o] * h + ws[OFF_AGGB + o];
  }
}

__global__ __launch_bounds__(64) void k_scan3(const float* __restrict__ dtw,
    const float* __restrict__ dtb, const float* __restrict__ alog,
    const float* __restrict__ Dp, float* __restrict__ ws) {
  int blk = blockIdx.x;
  int dir = blk >> 9, b = (blk >> 5) & 15, ch = blk & 31;
  int d = threadIdx.x;
  const float* ut = ws + OFF_UT;
  const float* xd = ws + OFF_XDBL + ((size_t)(dir * 16 + b) * L) * 8;
  float* y = ws + OFF_Y + ((size_t)(dir * 16 + b)) * L * 64;
  float4 w = *(const float4*)&dtw[(dir * 64 + d) * 4];
  float bia = dtb[dir * 64 + d];
  float A = -__expf(alog[dir * 64 + d]);
  float Dd = Dp[dir * 64 + d];
  float h = ws[OFF_H0 + ((size_t)(dir * 16 + b) * NCHUNK + ch) * 64 + d];
  for (int i = 0; i < CHUNK; ++i) {
    int l = ch * CHUNK + i;
    int row = dir ? ((l & 63) * 64 + (l >> 6)) : l;
    float u = ut[((size_t)b * L + row) * 64 + d];
    const float* p = xd + (size_t)l * 8;
    float delta = softplusf(p[0] * w.x + p[1] * w.y + p[2] * w.z + p[3] * w.w + bia);
    float a = __expf(delta * A);
    float bb = delta * p[4] * u;
    h = a * h + bb;
    y[(size_t)row * 64 + d] = h * p[5] + Dd * u;   // row is row-major pixel for both dirs
  }
}

// -------- WMMA: out-projection, gating, base_scale*g + up -> d_out ch 0..63 --------
__global__ __launch_bounds__(64) void k_proj_out(const float* __restrict__ ow,
    const float* __restrict__ base_scale, float* __restrict__ ws, float* __restrict__ out) {
  __shared__ float s_w[64 * 64];
  __shared__ float s_y[16 * 66];   // [n][d]
  int t = threadIdx.x;
  int b = blockIdx.x >> 5, tb = blockIdx.x & 31;
  for (int i = t; i < 4096; i += 64) s_w[i] = ow[i];
  int lane = t & 31, wave = t >> 5;
  int n = lane & 15, koff = (lane >> 4) << 1, m = lane & 15, rhi = (lane >> 4) << 3;
  const float* y0 = ws + OFF_Y + ((size_t)b) * L * 64;
  const float* y1 = ws + OFF_Y + ((size_t)(16 + b)) * L * 64;
  const float* z  = ws + OFF_Z + ((size_t)b * 64) * L;
  const float* up = ws + OFF_UP + ((size_t)b * 64) * L;
  for (int tile = 0; tile < 8; ++tile) {
    int l0 = tb * 128 + tile * 16;
    __syncthreads();
    for (int i = 0; i < 16; ++i) {
      int d = i * 4 + (t >> 4), nn = t & 15;
      float ys = y0[(size_t)(l0 + nn) * 64 + d] + y1[(size_t)(l0 + nn) * 64 + d];
      float zv = z[(size_t)d * L + l0 + nn];
      s_y[nn * 66 + d] = ys * siluf(zv);
    }
    __syncthreads();
    v8f a0 = {}, a1 = {};
#pragma unroll
    for (int k = 0; k < 16; ++k) {
      int kk = k * 4 + koff;
      v2f bf = *(const v2f*)&s_y[n * 66 + kk];
      v2f w0 = *(const v2f*)&s_w[((wave * 2 + 0) * 16 + m) * 64 + kk];
      v2f w1 = *(const v2f*)&s_w[((wave * 2 + 1) * 16 + m) * 64 + kk];
      a0 = wmma4(w0, bf, a0);
      a1 = wmma4(w1, bf, a1);
    }
    int l = l0 + n;
#pragma unroll
    for (int mt = 0; mt < 2; ++mt) {
      v8f a = mt ? a1 : a0;
      int ob = (wave * 2 + mt) * 16;
#pragma unroll
      for (int r = 0; r < 8; ++r) {
        int o = ob + r + rhi;
        out[((size_t)(b * Ctot + o)) * L + l] = base_scale[o] * a[r] + up[(size_t)o * L + l];
      }
    }
  }
}

extern "C" void kernel_launch(void* const* d_in, const int* in_sizes, int n_in,
                              void* d_out, int out_size, void* d_ws, size_t ws_size,
                              hipStream_t stream) {
  (void)in_sizes; (void)n_in; (void)out_size; (void)ws_size;
  const float* x = (const float*)d_in[0];
  const float* ew[3]   = {(const float*)d_in[1],  (const float*)d_in[8],  (const float*)d_in[15]};
  const float* gw[3]   = {(const float*)d_in[2],  (const float*)d_in[9],  (const float*)d_in[16]};
  const float* bn1g[3] = {(const float*)d_in[3],  (const float*)d_in[10], (const float*)d_in[17]};
  const float* bn1b[3] = {(const float*)d_in[4],  (const float*)d_in[11], (const float*)d_in[18]};
  const float* pw[3]   = {(const float*)d_in[5],  (const float*)d_in[12], (const float*)d_in[19]};
  const float* bn2g[3] = {(const float*)d_in[6],  (const float*)d_in[13], (const float*)d_in[20]};
  const float* bn2b[3] = {(const float*)d_in[7],  (const float*)d_in[14], (const float*)d_in[21]};
  const float* wav_w     = (const float*)d_in[22];
  const float* wav_b     = (const float*)d_in[23];
  const float* wav_scale = (const float*)d_in[24];
  const float* base_sc   = (const float*)d_in[25];
  const float* in_w      = (const float*)d_in[26];
  const float* conv_w    = (const float*)d_in[27];
  const float* conv_b    = (const float*)d_in[28];
  const float* xproj     = (const float*)d_in[29];
  const float* dtw       = (const float*)d_in[30];
  const float* dtb       = (const float*)d_in[31];
  const float* alog      = (const float*)d_in[32];
  const float* Dp        = (const float*)d_in[33];
  const float* ow        = (const float*)d_in[34];
  float* out = (float*)d_out;
  float* ws  = (float*)d_ws;

  k_copy<<<8192, 256, 0, stream>>>((const float4*)x, (float4*)out);
  k_mean<<<1024, 256, 0, stream>>>(x, ws);
  k_gates_eff<<<48, 64, 0, stream>>>(ws, gw[0], gw[1], gw[2], ew[0], ew[1], ew[2]);
  k_local<<<1024, 256, 0, stream>>>(x, ws,
      bn1g[0], bn1b[0], pw[0], bn2g[0], bn2b[0],
      bn1g[1], bn1b[1], pw[1], bn2g[1], bn2b[1],
      bn1g[2], bn1b[2], pw[2], bn2g[2], bn2b[2], out);
  k_wavelet<<<1024, 256, 0, stream>>>(x, wav_w, wav_b, wav_scale, ws);
  k_proj_in<<<512, 64, 0, stream>>>(x, in_w, ws);
  k_conv_xc<<<1024, 256, 0, stream>>>(conv_w, conv_b, ws);
  k_transpose0<<<1024, 256, 0, stream>>>(xproj, ws);
  k_scan1<<<1024, 64, 0, stream>>>(dtw, dtb, alog, ws);
  k_scan2<<<32, 64, 0, stream>>>(ws);
  k_scan3<<<1024, 64, 0, stream>>>(dtw, dtb, alog, Dp, ws);
  k_proj_out<<<512, 64, 0, stream>>>(ow, base_sc, ws, out);
}